// SwitchGate_48773648614357
// MI455X (gfx1250) — compile-verified
//
#include <hip/hip_runtime.h>
#include <hip/hip_bf16.h>

// Fused Switch-MoE gate for MI455X (gfx1250, wave32).
//  logits = X @ W^T + b  -> softmax(E=64) -> top2 mask -> / sum_over_B -> *capacity
// GEMM uses V_WMMA_F32_16X16X4_F32 (exact f32 math). X (268 MB) streamed once
// (HBM-bound, ~12us floor @ 23.3 TB/s); W (1 MB) stays L2-resident.
// K loop processes 16-K blocks: 10x global_load_b128 per block (permuted-K
// operand packing), explicitly pipelined one block ahead with a scheduling
// fence so the 16 WMMAs of block k overlap the loads of block k+1.

typedef __attribute__((ext_vector_type(2))) float v2f;
typedef __attribute__((ext_vector_type(4))) float v4f;
typedef __attribute__((ext_vector_type(8))) float v8f;

#define TOKENS_PER_WG 16
#define NB            4      // batch (fixed by reference setup)
#define NEXPERTS      64
#define KSLICES       4      // K split across waves for parallelism
#define EPAD          65     // padded expert stride (bank-conflict free)

#define WMMA_F32(A, B, C) \
    __builtin_amdgcn_wmma_f32_16x16x4_f32(false, (A), false, (B), (short)0, (C), false, false)

// One 16-K block of operands: per lane 8 contiguous floats (two b128 loads)
// for A and each of the 4 B n-tiles.
struct Blk {
    v4f aL, aH, b0L, b0H, b1L, b1H, b2L, b2H, b3L, b3H;
};

__device__ __forceinline__ Blk load_blk(const float* __restrict__ bx,
                                        const float* __restrict__ bw,
                                        int oA, int o0, int o1, int o2, int o3,
                                        int k) {
    Blk t;
    t.aL  = *(const v4f*)(bx + oA + k);  t.aH  = *(const v4f*)(bx + oA + k + 4);
    t.b0L = *(const v4f*)(bw + o0 + k);  t.b0H = *(const v4f*)(bw + o0 + k + 4);
    t.b1L = *(const v4f*)(bw + o1 + k);  t.b1H = *(const v4f*)(bw + o1 + k + 4);
    t.b2L = *(const v4f*)(bw + o2 + k);  t.b2H = *(const v4f*)(bw + o2 + k + 4);
    t.b3L = *(const v4f*)(bw + o3 + k);  t.b3H = *(const v4f*)(bw + o3 + k + 4);
    return t;
}

__device__ __forceinline__ void mm_blk(const Blk& t,
                                       v8f& acc0, v8f& acc1, v8f& acc2, v8f& acc3) {
    // 4 WMMA steps; A and B use the identical per-lane K permutation, so the
    // accumulated dot products are exact.
    acc0 = WMMA_F32(t.aL.lo, t.b0L.lo, acc0);
    acc1 = WMMA_F32(t.aL.lo, t.b1L.lo, acc1);
    acc2 = WMMA_F32(t.aL.lo, t.b2L.lo, acc2);
    acc3 = WMMA_F32(t.aL.lo, t.b3L.lo, acc3);

    acc0 = WMMA_F32(t.aL.hi, t.b0L.hi, acc0);
    acc1 = WMMA_F32(t.aL.hi, t.b1L.hi, acc1);
    acc2 = WMMA_F32(t.aL.hi, t.b2L.hi, acc2);
    acc3 = WMMA_F32(t.aL.hi, t.b3L.hi, acc3);

    acc0 = WMMA_F32(t.aH.lo, t.b0H.lo, acc0);
    acc1 = WMMA_F32(t.aH.lo, t.b1H.lo, acc1);
    acc2 = WMMA_F32(t.aH.lo, t.b2H.lo, acc2);
    acc3 = WMMA_F32(t.aH.lo, t.b3H.lo, acc3);

    acc0 = WMMA_F32(t.aH.hi, t.b0H.hi, acc0);
    acc1 = WMMA_F32(t.aH.hi, t.b1H.hi, acc1);
    acc2 = WMMA_F32(t.aH.hi, t.b2H.hi, acc2);
    acc3 = WMMA_F32(t.aH.hi, t.b3H.hi, acc3);
}

__global__ __launch_bounds__(512)
void switch_gate_wmma_kernel(const float* __restrict__ X,
                             const float* __restrict__ W,
                             const float* __restrict__ bias,
                             float* __restrict__ out,
                             int S, int D)
{
    __shared__ float s_row[NB * TOKENS_PER_WG * EPAD];   // logits, then masked scores

    const int tid  = threadIdx.x;
    const int wave = tid >> 5;
    const int lane = tid & 31;
    const int b    = wave & 3;     // batch handled by this wave
    const int ks   = wave >> 2;    // K-slice handled by this wave
    const int half = lane >> 4;    // 0: lanes 0-15, 1: lanes 16-31
    const int ln   = lane & 15;
    const int s0   = blockIdx.x * TOKENS_PER_WG;

    // ---------------- GEMM: 16 tokens x 64 experts, K-slice per wave ---------
    const int kchunk = D / KSLICES;
    const int kbeg   = ks * kchunk;

    // Wave-uniform scalar bases + fixed per-lane 32-bit offsets.
    const float* baseX = X + kbeg;
    const float* baseW = W + kbeg;
    // Per 16-K block, lane-half h covers K = [k + 8h, k + 8h + 8).
    const int offA  = (b * S + s0 + ln) * D + 8 * half;
    const int offB0 = ln * D + 8 * half;
    const int offB1 = offB0 + 16 * D;
    const int offB2 = offB0 + 32 * D;
    const int offB3 = offB0 + 48 * D;

    v8f acc0 = {}, acc1 = {}, acc2 = {}, acc3 = {};

    Blk cur = load_blk(baseX, baseW, offA, offB0, offB1, offB2, offB3, 0);

    #pragma unroll 2
    for (int k = 16; k < kchunk; k += 16) {
        // Issue next block's 10 x b128 loads.
        Blk nxt = load_blk(baseX, baseW, offA, offB0, offB1, offB2, offB3, k);
        // Pull the HBM-bound X stream further ahead (counter-free prefetch).
        __builtin_prefetch((const void*)(baseX + offA + k + 256), 0, 1);
        // Fence: keep the loads above the compute so the WMMA group only
        // waits for the PREVIOUS block's loads (loadcnt <= 10, not 0).
        __builtin_amdgcn_sched_barrier(0);
        // 16 WMMAs on the current block while next block is in flight.
        mm_blk(cur, acc0, acc1, acc2, acc3);
        cur = nxt;
    }
    // Drain last pipeline stage.
    mm_blk(cur, acc0, acc1, acc2, acc3);

    // ------------- deterministic cross-K-slice reduction into LDS ------------
    // C/D layout: VGPR r -> token (r + 8*half), expert column ln within n-tile.
    for (int phase = 0; phase < KSLICES; ++phase) {
        if (ks == phase) {
            #pragma unroll
            for (int r = 0; r < 8; ++r) {
                const int t = r + 8 * half;
                float* row = &s_row[(b * TOKENS_PER_WG + t) * EPAD];
                if (phase == 0) {
                    row[ 0 + ln] = acc0[r];
                    row[16 + ln] = acc1[r];
                    row[32 + ln] = acc2[r];
                    row[48 + ln] = acc3[r];
                } else {
                    row[ 0 + ln] += acc0[r];
                    row[16 + ln] += acc1[r];
                    row[32 + ln] += acc2[r];
                    row[48 + ln] += acc3[r];
                }
            }
        }
        __syncthreads();   // fixed order -> bitwise deterministic sums
    }

    // ------------- softmax + top-2 mask per (batch, token) row --------------
    if (tid < NB * TOKENS_PER_WG) {
        const int bb = tid >> 4;
        const int t  = tid & 15;
        float* row = &s_row[(bb * TOKENS_PER_WG + t) * EPAD];

        // pass 1: top-2 logits (softmax is monotonic; ties -> lowest index)
        float v1 = -1e38f, v2 = -1e38f;
        int   i1 = 0,      i2 = 0;
        for (int e = 0; e < NEXPERTS; ++e) {
            const float l = row[e] + bias[e];
            if (l > v1)      { v2 = v1; i2 = i1; v1 = l; i1 = e; }
            else if (l > v2) { v2 = l;  i2 = e; }
        }
        // pass 2: softmax denominator
        float sum = 0.0f;
        for (int e = 0; e < NEXPERTS; ++e)
            sum += expf(row[e] + bias[e] - v1);
        const float g1 = 1.0f / sum;             // expf(v1 - v1) / sum
        const float g2 = expf(v2 - v1) / sum;

        // overwrite row with masked gate scores (only top-2 nonzero)
        for (int e = 0; e < NEXPERTS; ++e) row[e] = 0.0f;
        row[i1] = g1;
        row[i2] = g2;
    }
    __syncthreads();

    // ------------- cross-batch normalization + output -----------------------
    const float capacity = 5.0f;   // int(1.25 * B), B = 4
    const float eps      = 1e-6f;
    for (int i = tid; i < NB * TOKENS_PER_WG * NEXPERTS; i += 512) {
        const int e  = i & (NEXPERTS - 1);
        const int t  = (i >> 6) & (TOKENS_PER_WG - 1);
        const int bb = i >> 10;
        float denom = eps;
        #pragma unroll
        for (int b2 = 0; b2 < NB; ++b2)
            denom += s_row[(b2 * TOKENS_PER_WG + t) * EPAD + e];
        const float v = s_row[(bb * TOKENS_PER_WG + t) * EPAD + e];
        out[(size_t)(bb * S + s0 + t) * NEXPERTS + e] = v / denom * capacity;
    }
}

extern "C" void kernel_launch(void* const* d_in, const int* in_sizes, int n_in,
                              void* d_out, int out_size, void* d_ws, size_t ws_size,
                              hipStream_t stream) {
    const float* X    = (const float*)d_in[0];   // [B,S,D] f32
    const float* W    = (const float*)d_in[1];   // [E,D]   f32
    const float* bias = (const float*)d_in[2];   // [E]     f32
    float* out        = (float*)d_out;           // [B,S,E] f32

    const int E = in_sizes[2];              // 64
    const int D = in_sizes[1] / E;          // 4096
    const int B = NB;                       // 4 (reference setup)
    const int S = in_sizes[0] / (B * D);    // 4096
    (void)E; (void)d_ws; (void)ws_size; (void)n_in; (void)out_size;

    dim3 grid(S / TOKENS_PER_WG);
    dim3 block(512);
    hipLaunchKernelGGL(switch_gate_wmma_kernel, grid, block, 0, stream,
                       X, W, bias, out, S, D);
}